// DeformablePatchSampler2d_61375082659938
// MI455X (gfx1250) — compile-verified
//
#include <hip/hip_runtime.h>

#define DEV_AS1 __attribute__((address_space(1)))
#define DEV_AS3 __attribute__((address_space(3)))

namespace {
constexpr int N_ = 8;
constexpr int C_ = 64;
constexpr int H_ = 384;
constexpr int W_ = 384;
constexpr int PATCH_ = 16;
constexpr int M_ = 64;          // NPH*NPW = 8*8
constexpr int LDSW = 19;        // padded window row stride (odd -> bank-conflict free)
constexpr int WCAP = LDSW * LDSW; // 361 floats per channel window (max needed 18x18)
constexpr int CPC = 8;          // channels per chunk
constexpr int CHUNKS = 4;       // chunks per block -> 32 channels per block
constexpr int ZBLK = C_ / (CPC * CHUNKS); // grid.z = 2
}

__device__ __forceinline__ int imin_(int a, int b) { return a < b ? a : b; }
__device__ __forceinline__ int imax_(int a, int b) { return a > b ? a : b; }
__device__ __forceinline__ int iclamp_(int v, int lo, int hi) {
  return imin_(imax_(v, lo), hi);
}

// ---- CDNA5 async global->LDS copy (ASYNCcnt-tracked, no VGPR round-trip) ----
// cpol = 1 -> TH=NT (non-temporal): staged lines are single-use in L2.
__device__ __forceinline__ void async_ld_f32(float* lds_dst, const float* gsrc) {
#if defined(__AMDGCN__) && __has_builtin(__builtin_amdgcn_global_load_async_to_lds_b32)
  __builtin_amdgcn_global_load_async_to_lds_b32(
      (DEV_AS1 int*)gsrc, (DEV_AS3 int*)lds_dst, 0, 1);
#elif defined(__AMDGCN__)
  DEV_AS3 float* l3 = (DEV_AS3 float*)lds_dst;
  asm volatile("global_load_async_to_lds_b32 %0, %1, off"
               :
               : "v"(l3), "v"(gsrc)
               : "memory");
#else
  *lds_dst = *gsrc; // host-pass placeholder (never codegen'd for device)
#endif
}

__device__ __forceinline__ void wait_async0() {
#if defined(__AMDGCN__) && __has_builtin(__builtin_amdgcn_s_wait_asynccnt)
  __builtin_amdgcn_s_wait_asynccnt(0);
#elif defined(__AMDGCN__)
  asm volatile("s_wait_asynccnt 0x0" ::: "memory");
#endif
}

__global__ __launch_bounds__(256) void deform_patch_sampler_kernel(
    const float* __restrict__ x,     // (N, C, H, W)
    const float* __restrict__ off,   // (M, 2)
    float* __restrict__ out)         // (N, M, C, 16, 16)
{
  __shared__ float smem[2][CPC][WCAP];

  const int m    = blockIdx.x;  // patch index
  const int n    = blockIdx.y;  // batch index
  const int cgrp = blockIdx.z;  // channel group (32 channels)
  const int tid  = threadIdx.x; // 0..255
  const int ph   = tid >> 4;    // output row within patch (drives x/column coord!)
  const int pw   = tid & 15;    // output col within patch (drives y/row coord!)

  const float off0 = off[2 * m + 0];
  const float off1 = off[2 * m + 1];

  // centers: ch = linspace(0, H, 12)[2:10] -> step*(i+2);  m = i*NPW + j
  const int mi = m >> 3;
  const int mj = m & 7;
  const float step = 384.0f / 11.0f;
  const float chc = step * (float)(mi + 2); // h-center -> feeds ix (column)
  const float cwc = step * (float)(mj + 2); // w-center -> feeds iy (row)

  auto ix_of = [&](int p) -> float {  // depends on ph
    float hc = chc + (float)(p - PATCH_ / 2);
    float gx = 2.0f * hc / (float)(H_ - 1) - 1.0f + off0;
    return ((gx + 1.0f) * (float)W_ - 1.0f) * 0.5f;
  };
  auto iy_of = [&](int p) -> float {  // depends on pw
    float wc = cwc + (float)(p - PATCH_ / 2);
    float gy = 2.0f * wc / (float)(W_ - 1) - 1.0f + off1;
    return ((gy + 1.0f) * (float)H_ - 1.0f) * 0.5f;
  };

  // ---- window bounds (uniform across block; coords are affine -> extremes at ends)
  const int ya = (int)floorf(iy_of(0)), yb = (int)floorf(iy_of(PATCH_ - 1));
  const int xa = (int)floorf(ix_of(0)), xb = (int)floorf(ix_of(PATCH_ - 1));
  const int rlo = iclamp_(imin_(ya, yb), 0, H_ - 1);
  const int rhi = iclamp_(imax_(ya, yb) + 1, 0, H_ - 1);
  const int clo = iclamp_(imin_(xa, xb), 0, W_ - 1);
  const int chi = iclamp_(imax_(xa, xb) + 1, 0, W_ - 1);
  const int nr  = rhi - rlo + 1;  // <= 18
  const int ncw = chi - clo + 1;  // <= 18
  (void)ncw;

  // ---- per-thread bilinear taps (image row <- pw, image col <- ph)
  const float ixv = ix_of(ph);
  const float iyv = iy_of(pw);
  const float x0f = floorf(ixv), y0f = floorf(iyv);
  const float wx1 = ixv - x0f, wx0 = 1.0f - wx1;
  const float wy1 = iyv - y0f, wy0 = 1.0f - wy1;
  const int x0 = (int)x0f, y0 = (int)y0f;
  const int x1 = x0 + 1,   y1 = y0 + 1;
  const float mx0 = (x0 >= 0 && x0 < W_) ? 1.0f : 0.0f;
  const float mx1 = (x1 >= 0 && x1 < W_) ? 1.0f : 0.0f;
  const float my0 = (y0 >= 0 && y0 < H_) ? 1.0f : 0.0f;
  const float my1 = (y1 >= 0 && y1 < H_) ? 1.0f : 0.0f;
  const int cx0 = iclamp_(x0, 0, W_ - 1) - clo;
  const int cx1 = iclamp_(x1, 0, W_ - 1) - clo;
  const int ry0 = iclamp_(y0, 0, H_ - 1) - rlo;
  const int ry1 = iclamp_(y1, 0, H_ - 1) - rlo;
  const float w00 = wy0 * wx0 * my0 * mx0;
  const float w01 = wy0 * wx1 * my0 * mx1;
  const float w10 = wy1 * wx0 * my1 * mx0;
  const float w11 = wy1 * wx1 * my1 * mx1;
  const int i00 = ry0 * LDSW + cx0;
  const int i01 = ry0 * LDSW + cx1;
  const int i10 = ry1 * LDSW + cx0;
  const int i11 = ry1 * LDSW + cx1;

  const int cbase0 = cgrp * (CPC * CHUNKS);
  const int lane = tid & 31;   // column within window (ncw <= 18 active lanes)
  const int wrow = tid >> 5;   // wave index -> starting row (8 waves)

  // Row-mapped async staging: each wave streams whole contiguous rows;
  // no integer division, uniform trip counts per wave.
  auto load_chunk = [&](int k, int b) {
    const int cb = cbase0 + k * CPC;
    if (lane < ncw) {
      for (int cc = 0; cc < CPC; ++cc) {
        const float* plane =
            x + (((size_t)(n * C_ + cb + cc)) * H_ + rlo) * W_ + clo + lane;
        float* dst = &smem[b][cc][lane];
        for (int r = wrow; r < nr; r += 8) {
          async_ld_f32(dst + r * LDSW, plane + (size_t)r * W_);
        }
      }
    }
  };

  auto compute_chunk = [&](int k, int b) {
    const int cb = cbase0 + k * CPC;
    const size_t obase =
        (((size_t)n * M_ + m) * C_ + cb) * (PATCH_ * PATCH_) + tid;
    for (int cc = 0; cc < CPC; ++cc) {
      const float* wnd = &smem[b][cc][0];
      const float r = wnd[i00] * w00 + wnd[i01] * w01 +
                      wnd[i10] * w10 + wnd[i11] * w11;
      // write-once 32MB stream: keep it out of L2
      __builtin_nontemporal_store(r, &out[obase + (size_t)cc * (PATCH_ * PATCH_)]);
    }
  };

  // ---- double-buffered async pipeline: load k+1 while computing k
  load_chunk(0, 0);
  wait_async0();
  __syncthreads();
  for (int k = 0; k < CHUNKS; ++k) {
    if (k + 1 < CHUNKS) load_chunk(k + 1, (k + 1) & 1);
    compute_chunk(k, k & 1);
    wait_async0();
    __syncthreads();
  }
}

extern "C" void kernel_launch(void* const* d_in, const int* in_sizes, int n_in,
                              void* d_out, int out_size, void* d_ws, size_t ws_size,
                              hipStream_t stream) {
  (void)in_sizes; (void)n_in; (void)out_size; (void)d_ws; (void)ws_size;
  const float* x   = (const float*)d_in[0];  // (8, 64, 384, 384) f32
  const float* off = (const float*)d_in[1];  // (64, 2) f32
  float* out = (float*)d_out;                // (8, 64, 64, 16, 16) f32

  dim3 grid(M_, N_, ZBLK);   // (64, 8, 2)
  dim3 block(256, 1, 1);     // 8 wave32s
  deform_patch_sampler_kernel<<<grid, block, 0, stream>>>(x, off, out);
}